// CGCNNLayer_66537633349675
// MI455X (gfx1250) — compile-verified
//
#include <hip/hip_runtime.h>
#include <math.h>

typedef float v2f __attribute__((ext_vector_type(2)));
typedef float v8f __attribute__((ext_vector_type(8)));

#define HDIM 48
#define RDIM 40
#define GDIM 32
#define WAVES_PER_BLOCK 8
#define EDGES_PER_WAVE 16
#define EDGES_PER_BLOCK 128
#define TILES_PER_BLOCK 4
#define ACT_STRIDE 52              // 52*l16 mod 64 covers 16 distinct mult-of-4 -> conflict-free b64 reads
#define PI_OVER_CUTOFF 0.52359877559829887308f  // pi/6

// ---------------------------------------------------------------------------
// Kernel A: zero the scratch accumulators (agg[N*48] ++ tot[N])
// ---------------------------------------------------------------------------
__global__ void zero_kernel(float* __restrict__ p, size_t n) {
  size_t i = (size_t)blockIdx.x * blockDim.x + threadIdx.x;
  if (i < n) p[i] = 0.0f;
}

// ---------------------------------------------------------------------------
// Kernel B: precompute bilinear element-pair table
//   T[a,b] = emb[a] . (adaptW @ emb[b])   (119x119 — replaces per-edge 32x32)
// ---------------------------------------------------------------------------
__global__ __launch_bounds__(256) void pair_table_kernel(
    const float* __restrict__ emb, const float* __restrict__ adaptW,
    float* __restrict__ T, int nel) {
  int idx = blockIdx.x * blockDim.x + threadIdx.x;
  if (idx >= nel * nel) return;
  int a = idx / nel, b = idx % nel;
  const float* ea = emb + (size_t)a * GDIM;
  const float* eb = emb + (size_t)b * GDIM;
  float acc = 0.0f;
  #pragma unroll 4
  for (int k = 0; k < GDIM; ++k) {
    const float* wrow = adaptW + (size_t)k * GDIM;
    float t = 0.0f;
    #pragma unroll
    for (int l = 0; l < GDIM; ++l) t += wrow[l] * eb[l];
    acc += ea[k] * t;
  }
  T[idx] = acc;
}

// ---------------------------------------------------------------------------
// Kernel C: per-edge filter MLP via fp32 WMMA + gated message scatter.
// Each wave: 16 edges x 48 features. Each block: 8 waves = 128 edges/tile,
// 4 tiles per block (weights staged in LDS once per block).
//   GEMM1: [16x40]x[40x48]  (v_wmma_f32_16x16x4_f32, 10 K-steps x 3 N-tiles)
//   SiLU -> LDS transpose -> GEMM2: [16x48]x[48x48] (12 K-steps x 3 N-tiles)
//   epilogue: (+b2 +pairLogit) * h[dst] * (cw*mj), atomic scatter into agg[src]
// Weights live in LDS as K-pair-interleaved float2 so every B operand is a
// single aligned ds_load_b64 into an even VGPR pair (no v_mov marshalling).
// ---------------------------------------------------------------------------
__global__ __launch_bounds__(256) void edge_kernel(
    const float* __restrict__ h,
    const int* __restrict__ esrc, const int* __restrict__ edst,
    const float* __restrict__ edge_attr,
    const float* __restrict__ dist,
    const int* __restrict__ atom_z,
    const float* __restrict__ node_mult,
    const float* __restrict__ W1, const float* __restrict__ b1,
    const float* __restrict__ W2, const float* __restrict__ b2,
    const float* __restrict__ T, int nel,
    float* __restrict__ agg, float* __restrict__ tot,
    int E) {
  // sW1P[p*HDIM + n] = {W1[n][2p], W1[n][2p+1]}  (i.e. W1^T K-pairs)
  __shared__ v2f sW1P[(RDIM / 2) * HDIM];
  __shared__ v2f sW2P[(HDIM / 2) * HDIM];
  __shared__ float sB1[HDIM], sB2[HDIM];
  __shared__ __align__(16) float sAct[WAVES_PER_BLOCK][16 * ACT_STRIDE];
  __shared__ int   sSrc[WAVES_PER_BLOCK][16];
  __shared__ int   sDst[WAVES_PER_BLOCK][16];
  __shared__ float sScale[WAVES_PER_BLOCK][16];
  __shared__ float sLogit[WAVES_PER_BLOCK][16];

  const int tid = threadIdx.x;
  const int lane = tid & 31;
  const int wv = tid >> 5;
  const int half = lane >> 4;      // K-pair selector per ISA f32 A/B layouts
  const int l16 = lane & 15;

  // stage K-pair-interleaved transposed weights in LDS (once per block)
  for (int i = tid; i < (RDIM / 2) * HDIM; i += 256) {
    int p = i / HDIM, n = i % HDIM;
    const float* src = W1 + n * RDIM + 2 * p;
    v2f w; w.x = src[0]; w.y = src[1];
    sW1P[i] = w;
  }
  for (int i = tid; i < (HDIM / 2) * HDIM; i += 256) {
    int p = i / HDIM, n = i % HDIM;
    const float* src = W2 + n * HDIM + 2 * p;
    v2f w; w.x = src[0]; w.y = src[1];
    sW2P[i] = w;
  }
  if (tid < HDIM) { sB1[tid] = b1[tid]; sB2[tid] = b2[tid]; }

  const int tile0 = blockIdx.x * TILES_PER_BLOCK;

  for (int t = 0; t < TILES_PER_BLOCK; ++t) {
    const int e0 = (tile0 + t) * EDGES_PER_BLOCK;  // uniform across block
    if (e0 >= E) break;

    // per-edge metadata (cutoff weight, multiplicity, pair logit, endpoints)
    if (tid < EDGES_PER_BLOCK) {
      int eg = e0 + tid;
      bool valid = eg < E;
      int ec = valid ? eg : (E - 1);
      int s = esrc[ec];
      int d = edst[ec];
      float dd = dist[ec];
      float cw = (dd < 6.0f) ? 0.5f * (__cosf(dd * PI_OVER_CUTOFF) + 1.0f) : 0.0f;
      float mj = node_mult[d];
      int w = tid >> 4, m = tid & 15;
      sSrc[w][m] = s;
      sDst[w][m] = d;
      sScale[w][m] = valid ? cw * mj : 0.0f;   // invalid tail edges contribute 0
      sLogit[w][m] = T[atom_z[s] * nel + atom_z[d]];
      if (valid) unsafeAtomicAdd(&tot[s], mj);
    }
    __syncthreads();   // also covers weight staging on first iteration

    // ----- GEMM1: edge_attr tile (A, from global) x W1^T (B, from LDS) -----
    int aEdge = e0 + wv * EDGES_PER_WAVE + l16;
    if (aEdge >= E) aEdge = E - 1;   // clamped load; scale=0 kills contribution
    const float* xrow = edge_attr + (size_t)aEdge * RDIM;

    v8f acc0 = {}, acc1 = {}, acc2 = {};
    #pragma unroll
    for (int kk = 0; kk < RDIM / 4; ++kk) {
      const int pIdx = 2 * kk + half;                       // K-pair index
      v2f a = *(const v2f*)(xrow + 2 * pIdx);               // one global b64
      v2f b0 = sW1P[pIdx * HDIM + l16];                     // one ds b64 each
      v2f b1v = sW1P[pIdx * HDIM + 16 + l16];
      v2f b2v = sW1P[pIdx * HDIM + 32 + l16];
      acc0 = __builtin_amdgcn_wmma_f32_16x16x4_f32(false, a, false, b0,  (short)0, acc0, false, false);
      acc1 = __builtin_amdgcn_wmma_f32_16x16x4_f32(false, a, false, b1v, (short)0, acc1, false, false);
      acc2 = __builtin_amdgcn_wmma_f32_16x16x4_f32(false, a, false, b2v, (short)0, acc2, false, false);
    }

    // bias + SiLU, scatter D-layout (row=VGPR, col=lane) into LDS act tile
    {
      float bn0 = sB1[l16], bn1 = sB1[16 + l16], bn2 = sB1[32 + l16];
      float* actw = sAct[wv];
      #pragma unroll
      for (int v = 0; v < 8; ++v) {
        int m = v + 8 * half;
        float x0 = acc0[v] + bn0; x0 = x0 / (1.0f + __expf(-x0));
        float x1 = acc1[v] + bn1; x1 = x1 / (1.0f + __expf(-x1));
        float x2 = acc2[v] + bn2; x2 = x2 / (1.0f + __expf(-x2));
        actw[m * ACT_STRIDE + l16]      = x0;
        actw[m * ACT_STRIDE + 16 + l16] = x1;
        actw[m * ACT_STRIDE + 32 + l16] = x2;
      }
    }
    __syncthreads();

    // ----- GEMM2: act tile (A, from LDS) x W2^T (B, from LDS) -----
    v8f g0 = {}, g1 = {}, g2 = {};
    {
      const float* actw = sAct[wv];
      #pragma unroll
      for (int kk = 0; kk < HDIM / 4; ++kk) {
        const int pIdx = 2 * kk + half;
        v2f a = *(const v2f*)(actw + l16 * ACT_STRIDE + 2 * pIdx);  // ds b64
        v2f b0 = sW2P[pIdx * HDIM + l16];
        v2f b1v = sW2P[pIdx * HDIM + 16 + l16];
        v2f b2v = sW2P[pIdx * HDIM + 32 + l16];
        g0 = __builtin_amdgcn_wmma_f32_16x16x4_f32(false, a, false, b0,  (short)0, g0, false, false);
        g1 = __builtin_amdgcn_wmma_f32_16x16x4_f32(false, a, false, b1v, (short)0, g1, false, false);
        g2 = __builtin_amdgcn_wmma_f32_16x16x4_f32(false, a, false, b2v, (short)0, g2, false, false);
      }
    }

    // ----- epilogue: gate -> message -> atomic scatter-add by src -----
    {
      float bb0 = sB2[l16], bb1 = sB2[16 + l16], bb2 = sB2[32 + l16];
      #pragma unroll
      for (int v = 0; v < 8; ++v) {
        int m = v + 8 * half;
        int srcN = sSrc[wv][m];
        int dstN = sDst[wv][m];
        float scale = sScale[wv][m];
        float logit = sLogit[wv][m];
        const float* hrow = h + (size_t)dstN * HDIM;
        float* arow = agg + (size_t)srcN * HDIM;
        float m0 = (g0[v] + bb0 + logit) * hrow[l16]      * scale;
        float m1 = (g1[v] + bb1 + logit) * hrow[16 + l16] * scale;
        float m2 = (g2[v] + bb2 + logit) * hrow[32 + l16] * scale;
        unsafeAtomicAdd(&arow[l16], m0);
        unsafeAtomicAdd(&arow[16 + l16], m1);
        unsafeAtomicAdd(&arow[32 + l16], m2);
      }
    }
    __syncthreads();   // protect metadata/act tiles before next iteration
  }
}

// ---------------------------------------------------------------------------
// Kernel D: wave-per-node mean-normalize + residual + LayerNorm.
// Lane covers feature {lane} and (lane<16) feature {32+lane}; shfl reductions.
// ---------------------------------------------------------------------------
__global__ __launch_bounds__(256) void finalize_kernel(
    const float* __restrict__ h, const float* __restrict__ agg,
    const float* __restrict__ tot, const float* __restrict__ gamma,
    const float* __restrict__ beta, float* __restrict__ out, int N) {
  const int lane = threadIdx.x & 31;
  const int wv = threadIdx.x >> 5;
  const int node = blockIdx.x * WAVES_PER_BLOCK + wv;
  if (node >= N) return;

  const float inv = 1.0f / fmaxf(tot[node], 1e-8f);
  const float* hrow = h + (size_t)node * HDIM;
  const float* arow = agg + (size_t)node * HDIM;

  float x0 = hrow[lane] + arow[lane] * inv;
  float x1 = (lane < 16) ? (hrow[32 + lane] + arow[32 + lane] * inv) : 0.0f;

  float s = x0 + x1;
  #pragma unroll
  for (int off = 16; off >= 1; off >>= 1) s += __shfl_xor(s, off, 32);
  const float mu = s * (1.0f / HDIM);

  float d0 = x0 - mu;
  float d1 = (lane < 16) ? (x1 - mu) : 0.0f;
  float vs = d0 * d0 + d1 * d1;
  #pragma unroll
  for (int off = 16; off >= 1; off >>= 1) vs += __shfl_xor(vs, off, 32);
  const float rstd = rsqrtf(vs * (1.0f / HDIM) + 1e-5f);

  out[(size_t)node * HDIM + lane] = d0 * rstd * gamma[lane] + beta[lane];
  if (lane < 16)
    out[(size_t)node * HDIM + 32 + lane] = d1 * rstd * gamma[32 + lane] + beta[32 + lane];
}

// ---------------------------------------------------------------------------
extern "C" void kernel_launch(void* const* d_in, const int* in_sizes, int n_in,
                              void* d_out, int out_size, void* d_ws, size_t ws_size,
                              hipStream_t stream) {
  const float* h         = (const float*)d_in[0];
  const int*   edge_idx  = (const int*)d_in[1];
  const float* edge_attr = (const float*)d_in[2];
  const float* dist      = (const float*)d_in[3];
  const int*   atom_z    = (const int*)d_in[4];
  const float* node_mult = (const float*)d_in[5];
  // d_in[6] = batch (unused: single graph)
  const float* W1    = (const float*)d_in[7];
  const float* b1    = (const float*)d_in[8];
  const float* W2    = (const float*)d_in[9];
  const float* b2    = (const float*)d_in[10];
  const float* emb   = (const float*)d_in[11];
  const float* adaptW= (const float*)d_in[12];
  const float* gamma = (const float*)d_in[13];
  const float* beta  = (const float*)d_in[14];

  const int N   = in_sizes[0] / HDIM;
  const int E   = in_sizes[3];
  const int nel = in_sizes[11] / GDIM;

  // workspace: agg[N*48] | tot[N] | pairT[nel*nel]
  float* agg = (float*)d_ws;
  float* tot = agg + (size_t)N * HDIM;
  float* T   = tot + N;

  const int* esrc = edge_idx;
  const int* edst = edge_idx + E;

  const size_t zn = (size_t)N * HDIM + (size_t)N;
  zero_kernel<<<(int)((zn + 255) / 256), 256, 0, stream>>>(agg, zn);
  pair_table_kernel<<<(nel * nel + 255) / 256, 256, 0, stream>>>(emb, adaptW, T, nel);
  const int edgesPerBlock = EDGES_PER_BLOCK * TILES_PER_BLOCK;
  edge_kernel<<<(E + edgesPerBlock - 1) / edgesPerBlock, 256, 0, stream>>>(
      h, esrc, edst, edge_attr, dist, atom_z, node_mult,
      W1, b1, W2, b2, T, nel, agg, tot, E);
  finalize_kernel<<<(N + WAVES_PER_BLOCK - 1) / WAVES_PER_BLOCK, 256, 0, stream>>>(
      h, agg, tot, gamma, beta, (float*)d_out, N);
}